// ProofGNN_21199958573549
// MI455X (gfx1250) — compile-verified
//
#include <hip/hip_runtime.h>

typedef __attribute__((ext_vector_type(2))) float v2f;
typedef __attribute__((ext_vector_type(8))) float v8f;

#define DIM   128
#define PAD   132   // row stride in floats; (132*r + k) % 64 == (4r + k) % 64 -> conflict-free b64 reads
#define BROWS 64    // output rows per block

// ---------------- node embedding gather: x = entity_emb[entity] + type_emb[node_type]
__global__ void gather_embed_kernel(const int* __restrict__ entity,
                                    const int* __restrict__ ntype,
                                    const float* __restrict__ eemb,
                                    const float* __restrict__ temb,
                                    float* __restrict__ x, int n) {
    int t = blockIdx.x * blockDim.x + threadIdx.x;
    int node = t >> 5, q = t & 31;
    if (node >= n) return;
    const float4* ev = (const float4*)(eemb + (size_t)entity[node] * DIM);
    const float4* tv = (const float4*)(temb + (size_t)ntype[node] * DIM);
    float4 a = ev[q], b = tv[q];
    float4 r; r.x = a.x + b.x; r.y = a.y + b.y; r.z = a.z + b.z; r.w = a.w + b.w;
    ((float4*)(x + (size_t)node * DIM))[q] = r;
}

// ---------------- degree / normalization
__global__ void deg_init_kernel(float* __restrict__ deg, int n) {
    int t = blockIdx.x * blockDim.x + threadIdx.x;
    if (t < n) deg[t] = 1.0f;               // self-loop
}
__global__ void deg_count_kernel(const int* __restrict__ dst, float* __restrict__ deg, int ne) {
    int t = blockIdx.x * blockDim.x + threadIdx.x;
    if (t < ne) atomicAdd(&deg[dst[t]], 1.0f);
}
__global__ void rsqrt_kernel(float* __restrict__ d, int n) {
    int t = blockIdx.x * blockDim.x + threadIdx.x;
    if (t < n) d[t] = rsqrtf(d[t]);         // deg >= 1 always (self-loop)
}

// ---------------- WMMA f32 GEMM: C[n,ncols] = A[n,128] @ W[128,ncols] (+bias)
// Block: 256 threads / 8 waves, 64 rows x up-to-128 cols. W staged transposed+padded
// in LDS so the k-loop is pure ds_load_b64 + v_wmma_f32_16x16x4_f32, no branches.
// Each wave: 1 B-fragment reused across 4 M-subtiles (4 accumulator chains).
__global__ void __launch_bounds__(256)
gemm_wmma_kernel(const float* __restrict__ A,
                 const float* __restrict__ W,
                 const float* __restrict__ bias,   // may be null
                 float* __restrict__ C,
                 int n, int ncols, int relu_in) {
    extern __shared__ float smem[];
    const int ncols16 = (ncols + 15) & ~15;
    float* Wt = smem;                        // [ncols16][PAD], Wt[c*PAD + k] = W[k][c] (0-padded)
    float* As = smem + ncols16 * PAD;        // [BROWS][PAD]

    const int r0 = blockIdx.x * BROWS;

    // stage W transposed (coalesced global read, masked zero-pad once)
    for (int i = threadIdx.x; i < DIM * ncols16; i += blockDim.x) {
        int r = i / ncols16;
        int c = i - r * ncols16;
        Wt[c * PAD + r] = (c < ncols) ? W[r * ncols + c] : 0.0f;
    }
    // stage A tile (ReLU fused on load for layer 2)
    for (int i = threadIdx.x; i < BROWS * DIM; i += blockDim.x) {
        int r = i >> 7, c = i & 127;
        int gr = r0 + r;
        float v = (gr < n) ? A[(size_t)gr * DIM + c] : 0.0f;
        if (relu_in) v = fmaxf(v, 0.0f);
        As[r * PAD + c] = v;
    }
    __syncthreads();

    const int lane = threadIdx.x & 31;
    const int wv   = threadIdx.x >> 5;
    const int c0   = wv * 16;
    if (c0 >= ncols16) return;               // idle waves for narrow (classifier) GEMM
    const int half = lane >> 4;              // 0: K pair {0,1}, 1: K pair {2,3}
    const int l16  = lane & 15;              // A: M index / B,D: N index
    const int col  = c0 + l16;

    const float* bp = Wt + col * PAD;
    const float* a0p = As + (l16     ) * PAD;
    const float* a1p = As + (l16 + 16) * PAD;
    const float* a2p = As + (l16 + 32) * PAD;
    const float* a3p = As + (l16 + 48) * PAD;

    v8f acc0 = {}, acc1 = {}, acc2 = {}, acc3 = {};
    for (int k = 0; k < DIM; k += 4) {
        int ka = k + (half << 1);
        v2f b  = *(const v2f*)(bp  + ka);
        v2f a0 = *(const v2f*)(a0p + ka);
        v2f a1 = *(const v2f*)(a1p + ka);
        v2f a2 = *(const v2f*)(a2p + ka);
        v2f a3 = *(const v2f*)(a3p + ka);
        acc0 = __builtin_amdgcn_wmma_f32_16x16x4_f32(false, a0, false, b, (short)0, acc0, false, false);
        acc1 = __builtin_amdgcn_wmma_f32_16x16x4_f32(false, a1, false, b, (short)0, acc1, false, false);
        acc2 = __builtin_amdgcn_wmma_f32_16x16x4_f32(false, a2, false, b, (short)0, acc2, false, false);
        acc3 = __builtin_amdgcn_wmma_f32_16x16x4_f32(false, a3, false, b, (short)0, acc3, false, false);
    }

    const bool colok = col < ncols;
    const float bb = (bias != nullptr && colok) ? bias[col] : 0.0f;
    v8f accs[4] = {acc0, acc1, acc2, acc3};
    #pragma unroll
    for (int m = 0; m < 4; ++m) {
        #pragma unroll
        for (int j = 0; j < 8; ++j) {
            int r = r0 + m * 16 + (half << 3) + j;   // D: VGPR j -> M = j + 8*half
            if (colok && r < n) C[(size_t)r * ncols + col] = accs[m][j] + bb;
        }
    }
}

// ---------------- self-loop term + bias: out = xw * dis^2 + b
__global__ void agg_init_kernel(const float* __restrict__ xw,
                                const float* __restrict__ dis,
                                const float* __restrict__ bias,
                                float* __restrict__ out, int n) {
    int t = blockIdx.x * blockDim.x + threadIdx.x;
    int node = t >> 5, q = t & 31;
    if (node >= n) return;
    float s = dis[node];
    float ss = s * s;
    float4 v = ((const float4*)(xw + (size_t)node * DIM))[q];
    float4 b = ((const float4*)bias)[q];
    float4 r; r.x = fmaf(v.x, ss, b.x); r.y = fmaf(v.y, ss, b.y);
              r.z = fmaf(v.z, ss, b.z); r.w = fmaf(v.w, ss, b.w);
    ((float4*)(out + (size_t)node * DIM))[q] = r;
}

// ---------------- edge scatter: out[dst] += xw[src] * dis[src]*dis[dst]
// One wave per edge; lane q handles dims [4q, 4q+4). L2-resident atomics.
__global__ void edge_scatter_kernel(const int* __restrict__ ei,
                                    const float* __restrict__ xw,
                                    const float* __restrict__ dis,
                                    float* __restrict__ out, int ne) {
    int t = blockIdx.x * blockDim.x + threadIdx.x;
    int e = t >> 5, q = t & 31;
    if (e >= ne) return;
    int s = ei[e];
    int d = ei[ne + e];
    float nrm = dis[s] * dis[d];
    float4 v = ((const float4*)(xw + (size_t)s * DIM))[q];
    float* o = out + (size_t)d * DIM + q * 4;
    atomicAdd(o + 0, v.x * nrm);
    atomicAdd(o + 1, v.y * nrm);
    atomicAdd(o + 2, v.z * nrm);
    atomicAdd(o + 3, v.w * nrm);
}

extern "C" void kernel_launch(void* const* d_in, const int* in_sizes, int n_in,
                              void* d_out, int out_size, void* d_ws, size_t ws_size,
                              hipStream_t stream) {
    const int*   entity = (const int*)d_in[0];
    const int*   ntype  = (const int*)d_in[1];
    const int*   ei     = (const int*)d_in[2];
    const float* eemb   = (const float*)d_in[3];
    const float* temb   = (const float*)d_in[4];
    const float* W1     = (const float*)d_in[5];
    const float* b1     = (const float*)d_in[6];
    const float* W2     = (const float*)d_in[7];
    const float* b2     = (const float*)d_in[8];
    const float* Wc     = (const float*)d_in[9];
    const float* bc     = (const float*)d_in[10];

    int n    = in_sizes[0];
    int ne   = in_sizes[2] / 2;
    int ncls = in_sizes[10];

    float* bufA = (float*)d_ws;                 // ping
    float* bufB = bufA + (size_t)n * DIM;       // pong
    float* dis  = bufB + (size_t)n * DIM;       // [n]

    int nodeBlocks = (n * 32 + 255) / 256;
    int edgeBlocks = (int)(((long long)ne * 32 + 255) / 256);
    int tiles = (n + BROWS - 1) / BROWS;

    size_t shConv = (size_t)(128 * PAD + BROWS * PAD) * sizeof(float);   // ~101 KB (CDNA5: 320 KB/WG)
    int ncls16 = (ncls + 15) & ~15;
    size_t shCls  = (size_t)(ncls16 * PAD + BROWS * PAD) * sizeof(float);

    // x -> bufA
    gather_embed_kernel<<<nodeBlocks, 256, 0, stream>>>(entity, ntype, eemb, temb, bufA, n);
    // dis
    deg_init_kernel<<<(n + 255) / 256, 256, 0, stream>>>(dis, n);
    deg_count_kernel<<<(ne + 255) / 256, 256, 0, stream>>>(ei + ne, dis, ne);
    rsqrt_kernel<<<(n + 255) / 256, 256, 0, stream>>>(dis, n);

    // conv1: xw1 = x @ W1 -> bufB ; h1 = agg(xw1) + b1 -> bufA
    gemm_wmma_kernel<<<tiles, 256, shConv, stream>>>(bufA, W1, nullptr, bufB, n, DIM, 0);
    agg_init_kernel<<<nodeBlocks, 256, 0, stream>>>(bufB, dis, b1, bufA, n);
    edge_scatter_kernel<<<edgeBlocks, 256, 0, stream>>>(ei, bufB, dis, bufA, ne);

    // conv2: xw2 = relu(h1) @ W2 -> bufB ; h2 = agg(xw2) + b2 -> bufA
    gemm_wmma_kernel<<<tiles, 256, shConv, stream>>>(bufA, W2, nullptr, bufB, n, DIM, 1);
    agg_init_kernel<<<nodeBlocks, 256, 0, stream>>>(bufB, dis, b2, bufA, n);
    edge_scatter_kernel<<<edgeBlocks, 256, 0, stream>>>(ei, bufB, dis, bufA, ne);

    // classifier: logits = h2 @ Wc + bc -> d_out (wave 0 computes, cols 3..15 zero-padded)
    gemm_wmma_kernel<<<tiles, 256, shCls, stream>>>(bufA, Wc, bc, (float*)d_out, n, ncls, 0);
}